// improvehc_4234837754131
// MI455X (gfx1250) — compile-verified
//
#include <hip/hip_runtime.h>
#include <math.h>

// ---------------------------------------------------------------------------
// improve_hc objective on MI455X (gfx1250, wave32)
//
// Pipeline (all memory-bound; roofline ~75 MB @ 23.3 TB/s ≈ 4-5 us):
//   k_e1    : normalize + scale the gathered internal-node embedding
//   k_leaf  : per-leaf hyperbolic distance D[i] via V_WMMA_F32_16X16X4_F32
//             (16-leaf tiles staged in LDS, e1 replicated across B columns)
//   k_max   : block-partial max of D[l]+D[r] over pairs   (deterministic tree)
//   k_max2  : combine partials -> softmax shift
//   k_sum   : block-partial {sum e, sum sims*e, sum sims}
//   k_final : combine partials -> scalar output
// ---------------------------------------------------------------------------

typedef float v2f __attribute__((ext_vector_type(2)));
typedef float v8f __attribute__((ext_vector_type(8)));

#define RANK        128
#define INV_T       20.0f      // 1 / 0.05
#define MIN_SCALE   0.01f
#define MAX_SCALE   0.999f
#define EPSC        1e-7f
#define WAVES       4          // waves (=16-leaf tiles) per block in k_leaf
#define TSTRIDE     132        // padded LDS row stride (floats): 132%64==4 ->
                               // b64 A-operand reads hit all 64 banks
#define NB          1024       // grid for pair passes / partial array length
#define PB          256        // block size for pair passes

// ---- kernel 1: e1 = normalize(emb[lca]) * clip(scale), nx = ||e1||^2 -------
__global__ void k_e1(const int* __restrict__ lca, const float* __restrict__ emb,
                     const float* __restrict__ scale,
                     float* __restrict__ e1out, float* __restrict__ nxout) {
  __shared__ float red[RANK];
  const int t = threadIdx.x;
  const float e = emb[(size_t)lca[0] * RANK + t];
  red[t] = e * e;
  __syncthreads();
  for (int s = RANK / 2; s > 0; s >>= 1) {
    if (t < s) red[t] += red[t + s];
    __syncthreads();
  }
  const float norm2 = red[0];
  const float nrm   = sqrtf(norm2);
  const float sc    = fminf(fmaxf(scale[0], MIN_SCALE), MAX_SCALE);
  const float f     = sc / fmaxf(nrm, 1e-12f);
  e1out[t] = e * f;
  if (t == 0) nxout[0] = f * f * norm2;
}

// ---- kernel 2: D[i] = arccosh Poincare distance(e1, leaf_i) ----------------
// One wave handles a 16-leaf tile: LDS-staged A, WMMA f32 16x16x4, K=128.
__global__ void k_leaf(const float* __restrict__ leaves,
                       const float* __restrict__ e1g,
                       const float* __restrict__ nxg,
                       float* __restrict__ Dleaf,
                       int n_leaves, int ntiles) {
  __shared__ float e1s[RANK];
  __shared__ __align__(16) float tile[WAVES][16 * TSTRIDE];
  __shared__ float dsh[WAVES][16];

  const int tid  = threadIdx.x;
  const int w    = tid >> 5;
  const int lane = tid & 31;
  const int h    = lane >> 4;   // half-wave (K split for WMMA operands)
  const int r16  = lane & 15;

  if (tid < RANK) e1s[tid] = e1g[tid];
  __syncthreads();

  const float nx = nxg[0];
  const int t  = blockIdx.x * WAVES + w;
  const int tt = (t < ntiles) ? t : 0;       // keep control flow uniform
  const int leaf0 = tt * 16;

  // ---- stage 16 rows x 128 floats, coalesced 512B per row -----------------
#pragma unroll
  for (int j = 0; j < 16; ++j) {
    int row = leaf0 + j;
    row = (row < n_leaves) ? row : (n_leaves - 1);
    const float4 v = *(const float4*)(leaves + (size_t)row * RANK + lane * 4);
    *(float4*)&tile[w][j * TSTRIDE + lane * 4] = v;
  }
  __syncthreads();

  // ---- 32x V_WMMA_F32_16X16X4_F32, e1 replicated across all 16 B columns --
  // A layout (ISA 7.12.2): lane<16 holds (M=lane, K=kk+0/kk+1) in v0/v1,
  //                        lane>=16 holds (M=lane-16, K=kk+2/kk+3).
  // B layout mirrors: lane = column N, VGPR/half = K -> replicating e1 over N
  // makes every column of D equal the per-leaf dot products.
  v8f acc = {0.f, 0.f, 0.f, 0.f, 0.f, 0.f, 0.f, 0.f};
  const float* arow = &tile[w][r16 * TSTRIDE + 2 * h];
  const float* brow = &e1s[2 * h];
#pragma unroll
  for (int kk = 0; kk < RANK; kk += 4) {
    v2f a; a.x = arow[kk]; a.y = arow[kk + 1];
    v2f b; b.x = brow[kk]; b.y = brow[kk + 1];
    acc = __builtin_amdgcn_wmma_f32_16x16x4_f32(
        /*neg_a=*/false, a, /*neg_b=*/false, b,
        /*c_mod=*/(short)0, acc, /*reuse_a=*/false, /*reuse_b=*/false);
  }

  // ---- ||leaf||^2 on the VALU from the same LDS tile ----------------------
  float ssq = 0.f;
  const float* srow = &tile[w][r16 * TSTRIDE + 64 * h];
#pragma unroll
  for (int i = 0; i < 64; i += 4) {
    const float4 v = *(const float4*)(srow + i);
    ssq += v.x * v.x + v.y * v.y + v.z * v.z + v.w * v.w;
  }
  ssq += __shfl_xor(ssq, 16, 32);            // combine K halves (wave32)

  // D accumulator: lane L, VGPR r holds dot[leaf r + 8*(L>>4)] (all N equal).
  if (lane == 0) {
#pragma unroll
    for (int r = 0; r < 8; ++r) dsh[w][r] = acc[r];
  }
  if (lane == 16) {
#pragma unroll
    for (int r = 0; r < 8; ++r) dsh[w][8 + r] = acc[r];
  }
  __syncthreads();

  if (t < ntiles && lane < 16) {
    const int leaf = t * 16 + lane;
    if (leaf < n_leaves) {
      const float dot = dsh[w][lane];
      const float ny  = ssq;
      const float sq  = nx - 2.f * dot + ny;          // ||e1 - leaf||^2
      float arg = 1.f + 2.f * sq / ((1.f - nx) * (1.f - ny));
      arg = fmaxf(arg, 1.f + EPSC);
      Dleaf[leaf] = acoshf(arg);
    }
  }
}

// ---- kernel 3: per-block max of D[l]+D[r] ----------------------------------
__global__ void k_max(const float* __restrict__ D, const int* __restrict__ li,
                      const int* __restrict__ ri, int n,
                      float* __restrict__ pmax) {
  __shared__ float red[PB];
  float m = -1.f;  // distances are strictly positive
  for (int i = blockIdx.x * PB + threadIdx.x; i < n; i += gridDim.x * PB)
    m = fmaxf(m, D[li[i]] + D[ri[i]]);
  red[threadIdx.x] = m;
  __syncthreads();
  for (int s = PB / 2; s > 0; s >>= 1) {
    if (threadIdx.x < s)
      red[threadIdx.x] = fmaxf(red[threadIdx.x], red[threadIdx.x + s]);
    __syncthreads();
  }
  if (threadIdx.x == 0) pmax[blockIdx.x] = red[0];
}

__global__ void k_max2(const float* __restrict__ pmax, float* __restrict__ zmax) {
  __shared__ float red[NB];
  const int t = threadIdx.x;
  red[t] = pmax[t];
  __syncthreads();
  for (int s = NB / 2; s > 0; s >>= 1) {
    if (t < s) red[t] = fmaxf(red[t], red[t + s]);
    __syncthreads();
  }
  if (t == 0) zmax[0] = red[0];
}

// ---- kernel 4: per-block {sum e, sum sims*e, sum sims} ---------------------
__global__ void k_sum(const float* __restrict__ D, const int* __restrict__ li,
                      const int* __restrict__ ri, const float* __restrict__ sims,
                      int n, const float* __restrict__ zmax,
                      float* __restrict__ pe, float* __restrict__ pse,
                      float* __restrict__ ps) {
  __shared__ float r0[PB], r1[PB], r2[PB];
  const float dmax = zmax[0];
  float se = 0.f, sse = 0.f, ss = 0.f;
  for (int i = blockIdx.x * PB + threadIdx.x; i < n; i += gridDim.x * PB) {
    const float d = D[li[i]] + D[ri[i]];
    const float e = expf((d - dmax) * INV_T);
    const float s = sims[i];
    se += e; sse += s * e; ss += s;
  }
  r0[threadIdx.x] = se; r1[threadIdx.x] = sse; r2[threadIdx.x] = ss;
  __syncthreads();
  for (int s = PB / 2; s > 0; s >>= 1) {
    if (threadIdx.x < s) {
      r0[threadIdx.x] += r0[threadIdx.x + s];
      r1[threadIdx.x] += r1[threadIdx.x + s];
      r2[threadIdx.x] += r2[threadIdx.x + s];
    }
    __syncthreads();
  }
  if (threadIdx.x == 0) {
    pe[blockIdx.x] = r0[0]; pse[blockIdx.x] = r1[0]; ps[blockIdx.x] = r2[0];
  }
}

__global__ void k_final(const float* __restrict__ pe, const float* __restrict__ pse,
                        const float* __restrict__ ps, float* __restrict__ out) {
  __shared__ float r0[NB], r1[NB], r2[NB];
  const int t = threadIdx.x;
  r0[t] = pe[t]; r1[t] = pse[t]; r2[t] = ps[t];
  __syncthreads();
  for (int s = NB / 2; s > 0; s >>= 1) {
    if (t < s) { r0[t] += r0[t + s]; r1[t] += r1[t + s]; r2[t] += r2[t + s]; }
    __syncthreads();
  }
  if (t == 0) out[0] = r2[0] - r1[0] / r0[0];   // sum(sims) - w_ord
}

// ---------------------------------------------------------------------------
extern "C" void kernel_launch(void* const* d_in, const int* in_sizes, int n_in,
                              void* d_out, int out_size, void* d_ws, size_t ws_size,
                              hipStream_t stream) {
  const int*   lca    = (const int*)  d_in[0];
  const int*   li     = (const int*)  d_in[1];
  const int*   ri     = (const int*)  d_in[2];
  const float* sims   = (const float*)d_in[3];
  const float* emb    = (const float*)d_in[4];
  const float* leaves = (const float*)d_in[5];
  const float* scale  = (const float*)d_in[6];

  const int n_pairs  = in_sizes[1];
  const int n_leaves = in_sizes[5] / RANK;

  // workspace layout (floats)
  float* ws    = (float*)d_ws;
  float* e1    = ws;                        // 128
  float* nx    = ws + 128;                  // 1
  float* zmax  = ws + 129;                  // 1
  float* Dleaf = ws + 256;                  // n_leaves (padded to 16)
  float* pmax  = Dleaf + ((n_leaves + 15) & ~15);
  float* pe    = pmax + NB;
  float* pse   = pe   + NB;
  float* ps    = pse  + NB;

  k_e1<<<1, RANK, 0, stream>>>(lca, emb, scale, e1, nx);

  const int ntiles = (n_leaves + 15) / 16;
  const int nb2    = (ntiles + WAVES - 1) / WAVES;
  k_leaf<<<nb2, 32 * WAVES, 0, stream>>>(leaves, e1, nx, Dleaf, n_leaves, ntiles);

  k_max <<<NB, PB, 0, stream>>>(Dleaf, li, ri, n_pairs, pmax);
  k_max2<<<1, NB, 0, stream>>>(pmax, zmax);
  k_sum <<<NB, PB, 0, stream>>>(Dleaf, li, ri, sims, n_pairs, zmax, pe, pse, ps);
  k_final<<<1, NB, 0, stream>>>(pe, pse, ps, (float*)d_out);
}